// GNN_EdgeAttr_12360915878361
// MI455X (gfx1250) — compile-verified
//
#include <hip/hip_runtime.h>
#include <math.h>

typedef float v2f __attribute__((ext_vector_type(2)));
typedef float v8f __attribute__((ext_vector_type(8)));

#define NN 10000
#define NE 300000
#define MT 80   // M rows per wave (5 x 16); 10000 % 80 == 0

// ---------------- agg init: agg = x (h = agg + x fusion) ----------------
__global__ void copy_vec4(const float* __restrict__ s, float* __restrict__ d, int n4) {
    int i = blockIdx.x * blockDim.x + threadIdx.x;
    if (i < n4)
        reinterpret_cast<float4*>(d)[i] = reinterpret_cast<const float4*>(s)[i];
}

// ---------------- W repack: Wp[k/2][2n+j] = W[k+j][n], j in {0,1} ----------------
// makes each WMMA B-fragment {W[k][n], W[k+1][n]} a single aligned b64 load
__global__ void repack_w(const float* __restrict__ W, float* __restrict__ Wp, int K, int N) {
    int idx = blockIdx.x * blockDim.x + threadIdx.x;
    int total = (K >> 1) * N;
    if (idx >= total) return;
    int k2 = idx / N;
    int n  = idx - k2 * N;
    v2f p;
    p.x = W[(size_t)(2 * k2 + 0) * N + n];
    p.y = W[(size_t)(2 * k2 + 1) * N + n];
    *reinterpret_cast<v2f*>(Wp + (size_t)k2 * 2 * N + 2 * n) = p;
}

// ---------------- edge message + scatter-add ----------------
// one wave32 per edge; lane covers 4 floats, loop strides 128 features
__global__ void edge_scatter(const float* __restrict__ x,
                             const int* __restrict__ src,
                             const int* __restrict__ dst,
                             const float* __restrict__ eattr,
                             const float* __restrict__ ew,
                             const float* __restrict__ eb,
                             float* __restrict__ agg,
                             int nEdges, int dim) {
    int wid  = (blockIdx.x * blockDim.x + threadIdx.x) >> 5;
    int lane = threadIdx.x & 31;
    if (wid >= nEdges) return;
    int s = src[wid];
    int d = dst[wid];
    float a = eattr[wid];
    const float* xr = x + (size_t)s * dim;
    float* ar = agg + (size_t)d * dim;
    for (int f = lane * 4; f < dim; f += 128) {
        float4 xv = *reinterpret_cast<const float4*>(xr + f);
        float4 wv = *reinterpret_cast<const float4*>(ew + f);
        float4 bv = *reinterpret_cast<const float4*>(eb + f);
        float m0 = fmaxf(xv.x + a * wv.x + bv.x, 0.f);
        float m1 = fmaxf(xv.y + a * wv.y + bv.y, 0.f);
        float m2 = fmaxf(xv.z + a * wv.z + bv.z, 0.f);
        float m3 = fmaxf(xv.w + a * wv.w + bv.w, 0.f);
        atomicAdd(ar + f + 0, m0);
        atomicAdd(ar + f + 1, m1);
        atomicAdd(ar + f + 2, m2);
        atomicAdd(ar + f + 3, m3);
    }
}

// ---------------- fp32 WMMA GEMM: D = [relu](A @ W + bias) ----------------
// A: [M,K] row-major, Wp: pair-interleaved W (see repack_w), bias: [N].
// One wave computes an 80(M) x 64(N) tile = 5x4 grid of 16x16 WMMA tiles,
// K stepped by 4 with V_WMMA_F32_16X16X4_F32.
// __launch_bounds__(128, 1): min 1 wave/EU -> full VGPR budget, no acc spills
// (160 accumulator VGPRs + frags ~= 210 < 256).
// A frag i (16x4):  lane(0-15)=row m0+16i+l16, v = {A[m][k0+2h], A[m][k0+2h+1]}
// B frag j (4x16):  lane(0-15)=col n0+16j+l16, one b64 from Wp at pair p=k0/2+h
// C/D (16x16 f32):  VGPR r holds M = r + 8h, N = l16
template <bool RELU>
__global__ __launch_bounds__(128, 1) void
gemm_wmma_f32(const float* __restrict__ A,
              const float* __restrict__ Wp,
              const float* __restrict__ bias,
              float* __restrict__ D,
              int M, int K, int N) {
    const int lane = threadIdx.x & 31;
    const int gw   = (blockIdx.x * blockDim.x + threadIdx.x) >> 5;
    const int nt64   = N >> 6;
    const int mtiles = M / MT;
    if (gw >= mtiles * nt64) return;       // uniform per-wave: EXEC stays all-1s
    const int m0   = (gw / nt64) * MT;
    const int n0   = (gw % nt64) << 6;
    const int half = lane >> 4;
    const int l16  = lane & 15;

    v8f acc[5][4] = {};
    const float* arow[5];
#pragma unroll
    for (int i = 0; i < 5; ++i)
        arow[i] = A + (size_t)(m0 + 16 * i + l16) * K + 2 * half;
    const float* wbase = Wp + 2 * (n0 + l16);

    for (int k0 = 0; k0 < K; k0 += 4) {
        v2f a[5], b[4];
#pragma unroll
        for (int i = 0; i < 5; ++i)
            a[i] = *reinterpret_cast<const v2f*>(arow[i] + k0);
        const float* wrow = wbase + (size_t)((k0 >> 1) + half) * 2 * N;
#pragma unroll
        for (int j = 0; j < 4; ++j)
            b[j] = *reinterpret_cast<const v2f*>(wrow + 32 * j);
#pragma unroll
        for (int i = 0; i < 5; ++i)
#pragma unroll
            for (int j = 0; j < 4; ++j)
                acc[i][j] = __builtin_amdgcn_wmma_f32_16x16x4_f32(
                    false, a[i], false, b[j], (short)0, acc[i][j], false, false);
    }

    float bb[4];
#pragma unroll
    for (int j = 0; j < 4; ++j)
        bb[j] = bias[n0 + 16 * j + l16];
#pragma unroll
    for (int i = 0; i < 5; ++i) {
#pragma unroll
        for (int r = 0; r < 8; ++r) {
            const int m = m0 + 16 * i + half * 8 + r;
            float* drow = D + (size_t)m * N + n0 + l16;
#pragma unroll
            for (int j = 0; j < 4; ++j) {
                float v = acc[i][j][r] + bb[j];
                if (RELU) v = fmaxf(v, 0.f);
                drow[16 * j] = v;
            }
        }
    }
}

// ---------------- head: out = sigmoid(h @ fc2_w + fc2_b), OUT_DIM = 1 ----------------
__global__ void fc2_sigmoid(const float* __restrict__ h,
                            const float* __restrict__ w,
                            const float* __restrict__ b,
                            float* __restrict__ out, int M, int K) {
    int i = blockIdx.x * blockDim.x + threadIdx.x;
    if (i >= M) return;
    float acc = b[0];
    const float* hr = h + (size_t)i * K;
    for (int k = 0; k < K; k += 4) {
        float4 hv = *reinterpret_cast<const float4*>(hr + k);
        float4 wv = *reinterpret_cast<const float4*>(w + k);
        acc += hv.x * wv.x + hv.y * wv.y + hv.z * wv.z + hv.w * wv.w;
    }
    out[i] = 1.f / (1.f + __expf(-acc));
}

static inline void launch_gemm(const float* A, const float* W, const float* bias,
                               float* D, float* wpak, int M, int K, int N,
                               hipStream_t stream) {
    int tot = (K >> 1) * N;
    repack_w<<<(tot + 255) / 256, 256, 0, stream>>>(W, wpak, K, N);
    int waves = (M / MT) * (N >> 6);
    gemm_wmma_f32<true><<<(waves + 3) / 4, 128, 0, stream>>>(A, wpak, bias, D, M, K, N);
}

extern "C" void kernel_launch(void* const* d_in, const int* in_sizes, int n_in,
                              void* d_out, int out_size, void* d_ws, size_t ws_size,
                              hipStream_t stream) {
    (void)in_sizes; (void)n_in; (void)out_size; (void)ws_size;

    // setup_inputs() insertion order: x, edge_index, edge_attr, params
    // params: 3 convs x {edge_w, edge_b, w1, b1, w2, b2}, then fc1_w, fc1_b, fc2_w, fc2_b
    const float* x  = (const float*)d_in[0];
    const int*   ei = (const int*)d_in[1];     // [2, E] int32 (JAX x64 disabled)
    const float* ea = (const float*)d_in[2];
    const float* cp[3][6];
    for (int i = 0; i < 3; ++i)
        for (int j = 0; j < 6; ++j)
            cp[i][j] = (const float*)d_in[3 + i * 6 + j];
    const float* fc1_w = (const float*)d_in[21];
    const float* fc1_b = (const float*)d_in[22];
    const float* fc2_w = (const float*)d_in[23];
    const float* fc2_b = (const float*)d_in[24];

    const int* src = ei;
    const int* dst = ei + NE;

    // workspace: 3 x N*256 f32 activation buffers + 256KB weight-repack buffer
    float* bufA = (float*)d_ws;                 // agg
    float* bufB = bufA + (size_t)NN * 256;      // MLP hidden
    float* bufC = bufB + (size_t)NN * 256;      // layer output
    float* wpak = bufC + (size_t)NN * 256;      // repacked W (max 256*256 floats)

    const int dims_in[3]  = {128, 256, 256};
    const int dims_out[3] = {256, 256, 256};
    const float* xin = x;

    for (int L = 0; L < 3; ++L) {
        const int din  = dims_in[L];
        const int dout = dims_out[L];
        // agg = x  (fuses h = agg + x)
        {
            int n4 = NN * din / 4;
            copy_vec4<<<(n4 + 255) / 256, 256, 0, stream>>>(xin, bufA, n4);
        }
        // agg += scatter_add(relu(x[src] + edge_attr*ew + eb), dst)
        edge_scatter<<<(NE + 7) / 8, 256, 0, stream>>>(
            xin, src, dst, ea, cp[L][0], cp[L][1], bufA, NE, din);
        // hid = relu(agg @ w1 + b1)
        launch_gemm(bufA, cp[L][2], cp[L][3], bufB, wpak, NN, din, dout, stream);
        // x = relu(hid @ w2 + b2)   (outer relu of the reference fused in)
        launch_gemm(bufB, cp[L][4], cp[L][5], bufC, wpak, NN, dout, dout, stream);
        xin = bufC;
    }

    // head: h = relu(x @ fc1_w + fc1_b)  [256 -> 128]
    launch_gemm(bufC, fc1_w, fc1_b, bufA, wpak, NN, 256, 128, stream);
    // out = sigmoid(h @ fc2_w + fc2_b)  [128 -> 1]
    fc2_sigmoid<<<(NN + 255) / 256, 256, 0, stream>>>(
        bufA, fc2_w, fc2_b, (float*)d_out, NN, 128);
}